// ImiRegularizedWarper_9887014715993
// MI455X (gfx1250) — compile-verified
//
#include <hip/hip_runtime.h>

// Problem geometry (fixed by reference).
#define DD 160
#define HH 160
#define WW 160
#define BB 2
#define HWW (HH * WW)        // 25600
#define DHW (DD * HWW)       // 4,096,000
#define NV  (BB * DHW)       // 8,192,000 warp voxels  (== warped element count)
#define NG  (3 * NV)         // 24,576,000 gradient elements (mean denominator)
#define ALPHA_F 2.0f

#define NBLOCKS  2048
#define NTHREADS 256

typedef float v2f __attribute__((ext_vector_type(2)));
typedef float v8f __attribute__((ext_vector_type(8)));

// Fused kernel: trilinear backward warp + diffusive-regularizer partial sums.
__global__ __launch_bounds__(NTHREADS)
void warp_grad_kernel(const float* __restrict__ image,
                      const float* __restrict__ disp,
                      float* __restrict__ warped,
                      float* __restrict__ partials)
{
    __shared__ float sm[NTHREADS];
    const int tid = threadIdx.x;
    const int gsz = NBLOCKS * NTHREADS;
    float acc = 0.0f;

    for (int i = blockIdx.x * NTHREADS + tid; i < NV; i += gsz) {
        // i -> (b, z, y, x); warped flat index == i
        int x = i % WW;
        int t = i / WW;
        int y = t % HH;
        t /= HH;
        int z = t % DD;
        int b = t / DD;
        const int vi = z * HWW + y * WW + x;

        const float* __restrict__ db = disp  + (size_t)b * (3 * DHW);
        const float* __restrict__ ib = image + (size_t)b * DHW;

        // Warm the gather neighborhood: disp ~ N(0,1), so the 8 taps sit at/near
        // the identity cacheline.  Emits global_prefetch_b8 on gfx1250.
        __builtin_prefetch(ib + vi, 0, 3);

        const float d0 = db[vi];            // z displacement
        const float d1 = db[DHW + vi];      // y displacement
        const float d2 = db[2 * DHW + vi];  // x displacement

        // ---- trilinear warp (spacing == 1, border clamp) ----
        float cz = fminf(fmaxf((float)z + d0, 0.0f), (float)(DD - 1));
        float cy = fminf(fmaxf((float)y + d1, 0.0f), (float)(HH - 1));
        float cx = fminf(fmaxf((float)x + d2, 0.0f), (float)(WW - 1));
        float z0f = floorf(cz), y0f = floorf(cy), x0f = floorf(cx);
        int z0 = (int)z0f, y0 = (int)y0f, x0 = (int)x0f;
        int z1 = min(z0 + 1, DD - 1);
        int y1 = min(y0 + 1, HH - 1);
        int x1 = min(x0 + 1, WW - 1);
        float wz = cz - z0f, wy = cy - y0f, wx = cx - x0f;

        int zb0 = z0 * HWW, zb1 = z1 * HWW, yb0 = y0 * WW, yb1 = y1 * WW;
        float v000 = ib[zb0 + yb0 + x0], v001 = ib[zb0 + yb0 + x1];
        float v010 = ib[zb0 + yb1 + x0], v011 = ib[zb0 + yb1 + x1];
        float v100 = ib[zb1 + yb0 + x0], v101 = ib[zb1 + yb0 + x1];
        float v110 = ib[zb1 + yb1 + x0], v111 = ib[zb1 + yb1 + x1];

        float c00 = v000 + wx * (v001 - v000);
        float c01 = v010 + wx * (v011 - v010);
        float c10 = v100 + wx * (v101 - v100);
        float c11 = v110 + wx * (v111 - v110);
        float c0  = c00 + wy * (c01 - c00);
        float c1  = c10 + wy * (c11 - c10);
        float res = c0 + wz * (c1 - c0);
        __builtin_nontemporal_store(res, &warped[i]);  // streaming output

        // ---- diffusive regularizer: |grad u|^2, jnp.gradient semantics ----
        const float dc[3] = {d0, d1, d2};
        #pragma unroll
        for (int ch = 0; ch < 3; ++ch) {
            const float* __restrict__ p = db + ch * DHW;
            const float ctr = dc[ch];
            float gz = (z == 0)      ? p[vi + HWW] - ctr
                     : (z == DD - 1) ? ctr - p[vi - HWW]
                                     : 0.5f * (p[vi + HWW] - p[vi - HWW]);
            float gy = (y == 0)      ? p[vi + WW] - ctr
                     : (y == HH - 1) ? ctr - p[vi - WW]
                                     : 0.5f * (p[vi + WW] - p[vi - WW]);
            float gx = (x == 0)      ? p[vi + 1] - ctr
                     : (x == WW - 1) ? ctr - p[vi - 1]
                                     : 0.5f * (p[vi + 1] - p[vi - 1]);
            acc += gz * gz + gy * gy + gx * gx;
        }
    }

    // Deterministic block reduction: LDS tree then wave32 butterfly.
    sm[tid] = acc;
    __syncthreads();
    for (int off = NTHREADS / 2; off > 32; off >>= 1) {
        if (tid < off) sm[tid] += sm[tid + off];
        __syncthreads();
    }
    if (tid < 32) {
        float s = sm[tid] + sm[tid + 32];
        s += __shfl_xor(s, 16);
        s += __shfl_xor(s, 8);
        s += __shfl_xor(s, 4);
        s += __shfl_xor(s, 2);
        s += __shfl_xor(s, 1);
        if (tid == 0) partials[blockIdx.x] = s;
    }
}

// Final reduction of NBLOCKS partials -> loss scalar at d_out[NV].
// The last 32->1 step uses v_wmma_f32_16x16x4_f32: A holds the 32 lane values
// (A[m][0]=v[m], A[m][2]=v[m+16], rest 0), B = ones, so D[m][*] = v[m]+v[m+16];
// summing the 8 accumulator VGPRs gives rows 0-7 (lanes 0-15) / rows 8-15
// (lanes 16-31); one shfl_xor(16) completes the total.
__global__ __launch_bounds__(NTHREADS)
void loss_finalize_kernel(const float* __restrict__ partials,
                          float* __restrict__ out)
{
    __shared__ float sm[NTHREADS];
    const int tid = threadIdx.x;
    float s = 0.0f;
    for (int i = tid; i < NBLOCKS; i += NTHREADS) s += partials[i];  // fixed order
    sm[tid] = s;
    __syncthreads();
    for (int off = NTHREADS / 2; off > 32; off >>= 1) {
        if (tid < off) sm[tid] += sm[tid + off];
        __syncthreads();
    }
    if (tid < 32) {  // wave 0 only: EXEC all ones within the wave (wave32)
        float v = sm[tid] + sm[tid + 32];
        float total;
#if defined(__gfx1250__) && __has_builtin(__builtin_amdgcn_wmma_f32_16x16x4_f32)
        v2f a;  a[0] = v;    a[1] = 0.0f;   // 16x4 A: lanes 0-15 = K0/K1, 16-31 = K2/K3
        v2f bm; bm[0] = 1.0f; bm[1] = 1.0f; // 4x16 B = ones (layout-independent)
        v8f cacc = {0.f, 0.f, 0.f, 0.f, 0.f, 0.f, 0.f, 0.f};
        v8f d = __builtin_amdgcn_wmma_f32_16x16x4_f32(
            /*neg_a=*/false, a, /*neg_b=*/false, bm,
            /*c_mod=*/(short)0, cacc, /*reuse_a=*/false, /*reuse_b=*/false);
        float r = d[0] + d[1] + d[2] + d[3] + d[4] + d[5] + d[6] + d[7];
        total = r + __shfl_xor(r, 16);
#else
        float r = v;
        r += __shfl_xor(r, 16);
        r += __shfl_xor(r, 8);
        r += __shfl_xor(r, 4);
        r += __shfl_xor(r, 2);
        r += __shfl_xor(r, 1);
        total = r;
#endif
        if (tid == 0) out[NV] = ALPHA_F * total / (float)NG;
    }
}

extern "C" void kernel_launch(void* const* d_in, const int* in_sizes, int n_in,
                              void* d_out, int out_size, void* d_ws, size_t ws_size,
                              hipStream_t stream)
{
    const float* image = (const float*)d_in[0];  // [2,1,160,160,160] f32
    const float* disp  = (const float*)d_in[1];  // [2,3,160,160,160] f32
    float* out      = (float*)d_out;             // warped (NV floats) + loss (1)
    float* partials = (float*)d_ws;              // NBLOCKS floats of scratch

    warp_grad_kernel<<<NBLOCKS, NTHREADS, 0, stream>>>(image, disp, out, partials);
    loss_finalize_kernel<<<1, NTHREADS, 0, stream>>>(partials, out);
}